// CovariantGaugeAdapter_65515431133440
// MI455X (gfx1250) — compile-verified
//
#include <hip/hip_runtime.h>
#include <math.h>

typedef float v2f __attribute__((ext_vector_type(2)));
typedef float v8f __attribute__((ext_vector_type(8)));

#define BB 2
#define SS 512
#define DD 512
#define HH 8
#define HDIM 64
#define RR 16
#define INV_SQRT_HD 0.125f

// CDNA5 native transcendental tanh (gfx1250 V_TANH_F32). One independent
// op required after a TRANS op before its result is consumed -> v_nop.
__device__ __forceinline__ float fast_tanh(float x) {
  float y;
  asm volatile("v_tanh_f32 %0, %1\n\tv_nop" : "=v"(y) : "v"(x));
  return y;
}

// CDNA5 async copy: global memory -> LDS, 16B per lane, tracked by ASYNCcnt.
// LDS destination address = low 32 bits of the generic pointer (flat LDS
// aperture truncates to addr[31:0]).
__device__ __forceinline__ void async_copy_b128(const float* gsrc, float* ldst) {
  unsigned lds = (unsigned)(unsigned long long)(uintptr_t)ldst;
  unsigned long long ga = (unsigned long long)(uintptr_t)gsrc;
  asm volatile("global_load_async_to_lds_b128 %0, %1, off"
               :: "v"(lds), "v"(ga) : "memory");
}
__device__ __forceinline__ void wait_asynccnt0() {
  asm volatile("s_wait_asynccnt 0x0" ::: "memory");
}

// --- WMMA f32 16x16x4 fragment loaders (wave32) ---
// A (16x4, row-major rows in LDS, leading dim ldm):
//   lanes 0-15 hold row M=lane, VGPR0/1 = K,K+1 ; lanes 16-31 = K+2,K+3
__device__ __forceinline__ v2f frag_rowmajor(const float* sm, int lane, int k0, int ldm) {
  int hi = (lane >> 4) << 1;
  int r  = lane & 15;
  const float* p = sm + r * ldm + k0 + hi;
  v2f v; v[0] = p[0]; v[1] = p[1];
  return v;
}
// B (4x16) where B(k,n) = sm[(k0+k)*ldm + col0 + n] (k varies along rows)
__device__ __forceinline__ v2f frag_colmajor(const float* sm, int lane, int k0, int col0, int ldm) {
  int hi = (lane >> 4) << 1;
  int r  = lane & 15;
  v2f v;
  v[0] = sm[(k0 + hi)     * ldm + col0 + r];
  v[1] = sm[(k0 + hi + 1) * ldm + col0 + r];
  return v;
}

// ---------------- Kernel 1: LayerNorm + low-rank field MLP ----------------
__global__ void __launch_bounds__(256) prep_kernel(
    const float* __restrict__ hidden, const float* __restrict__ ln_g,
    const float* __restrict__ ln_b, const float* __restrict__ fg_w1,
    const float* __restrict__ fg_w2,
    float* __restrict__ Aq, float* __restrict__ Ak, float* __restrict__ Av) {
  __shared__ float xs[DD];
  __shared__ float red[256];
  __shared__ float tt[RR];
  const int row = blockIdx.x;            // b*S + s
  const int t = threadIdx.x;
  const float* xr = hidden + (size_t)row * DD;
  float a0 = xr[t], a1 = xr[t + 256];
  red[t] = a0 + a1;
  __syncthreads();
  for (int off = 128; off > 0; off >>= 1) { if (t < off) red[t] += red[t + off]; __syncthreads(); }
  float mean = red[0] * (1.0f / DD);
  __syncthreads();
  red[t] = a0 * a0 + a1 * a1;
  __syncthreads();
  for (int off = 128; off > 0; off >>= 1) { if (t < off) red[t] += red[t + off]; __syncthreads(); }
  float var = red[0] * (1.0f / DD) - mean * mean;
  float rstd = rsqrtf(var + 1e-5f);
  __syncthreads();
  xs[t]       = (a0 - mean) * rstd * ln_g[t]       + ln_b[t];
  xs[t + 256] = (a1 - mean) * rstd * ln_g[t + 256] + ln_b[t + 256];
  __syncthreads();
  if (t < RR) {
    const float* w = fg_w1 + t * DD;
    float dot = 0.f;
    for (int k = 0; k < DD; ++k) dot += xs[k] * w[k];
    tt[t] = dot / (1.0f + __expf(-dot));   // silu
  }
  __syncthreads();
  float tv[RR];
  #pragma unroll
  for (int r = 0; r < RR; ++r) tv[r] = tt[r];
  const size_t base = (size_t)row * DD;
  #pragma unroll
  for (int j = 0; j < 6; ++j) {
    int o = t + j * 256;                   // 0..1535
    const float* w = fg_w2 + o * RR;
    float v = 0.f;
    #pragma unroll
    for (int r = 0; r < RR; ++r) v += w[r] * tv[r];
    if (o < DD)            Aq[base + o] = v;
    else if (o < 2 * DD)   Ak[base + o - DD] = v;
    else                   Av[base + o - 2 * DD] = v;
  }
}

// --------- Kernel 2: per-head 64x64 transforms:  Q2 = ga*A_q@aq_w^T,  K' = k + ga*A_k@ak_w^T
__global__ void __launch_bounds__(256) headprep_kernel(
    const float* __restrict__ Aq, const float* __restrict__ Ak,
    const float* __restrict__ k_base,
    const float* __restrict__ aq_w, const float* __restrict__ ak_w,
    const float* __restrict__ g_attn,
    float* __restrict__ Qp2, float* __restrict__ Kp) {
  int idx = blockIdx.x * blockDim.x + threadIdx.x;   // over B*H*S*HD
  int d = idx & (HDIM - 1);
  int s = (idx >> 6) & (SS - 1);
  int h = (idx >> 15) & (HH - 1);
  int b = idx >> 18;
  const float* aqr = Aq + ((size_t)(b * SS + s)) * DD + h * HDIM;
  const float* akr = Ak + ((size_t)(b * SS + s)) * DD + h * HDIM;
  const float* wq = aq_w + d * HDIM;
  const float* wk = ak_w + d * HDIM;
  float sq = 0.f, sk = 0.f;
  #pragma unroll 8
  for (int e = 0; e < HDIM; ++e) { sq += wq[e] * aqr[e]; sk += wk[e] * akr[e]; }
  float ga = g_attn[h];
  Qp2[idx] = ga * sq;
  Kp[idx]  = k_base[idx] + ga * sk;
}

// --------- Kernel 3: fused flash attention with WMMA + tanh rel-bias -------
// K-side tiles staged with GLOBAL_LOAD_ASYNC_TO_LDS_B128 (ASYNCcnt) and
// double-buffered so tile kt+1 streams in while tile kt computes.
__global__ void __launch_bounds__(32) attn_kernel(
    const float* __restrict__ q_base, const float* __restrict__ k_base,
    const float* __restrict__ v_base,
    const float* __restrict__ Kp, const float* __restrict__ Qp2,
    const float* __restrict__ Aq, const float* __restrict__ Ak,
    const float* __restrict__ rel_bias, const float* __restrict__ g_rel,
    float* __restrict__ gauge) {
  __shared__ __align__(16) float qs[16 * HDIM], qps[16 * HDIM], aqs[16 * HDIM];
  __shared__ __align__(16) float ks[2 * 16 * HDIM], kbs[2 * 16 * HDIM];
  __shared__ __align__(16) float aks[2 * 16 * HDIM], vs[2 * 16 * HDIM];
  __shared__ float Ps[16 * 16];
  __shared__ float rb[HDIM];
  const int lane = threadIdx.x;
  const int qt = blockIdx.x & (SS / 16 - 1);
  const int h  = (blockIdx.x >> 5) & (HH - 1);
  const int b  = blockIdx.x >> 8;
  const int q0 = qt * 16;
  const size_t bh = ((size_t)b * HH + h) * SS;   // row base in [B,H,S,HD]
  const float grh = g_rel[h];
  for (int i = lane; i < 16 * HDIM; i += 32) {
    int r = i >> 6, c = i & 63;
    qs[i]  = q_base[(bh + q0 + r) * HDIM + c];
    qps[i] = Qp2  [(bh + q0 + r) * HDIM + c];
    aqs[i] = Aq   [((size_t)(b * SS + q0 + r)) * DD + h * HDIM + c];
  }
  rb[lane]      = rel_bias[h * HDIM + lane];
  rb[lane + 32] = rel_bias[h * HDIM + lane + 32];

  // async-stage one 4KB tile quartet into buffer `buf`
  auto issue_tile = [&](int k0, int buf) {
    const float* kp_g = Kp     + (bh + k0) * HDIM;   // tile-contiguous
    const float* kb_g = k_base + (bh + k0) * HDIM;
    const float* v_g  = v_base + (bh + k0) * HDIM;
    const float* ak_g = Ak + ((size_t)(b * SS + k0)) * DD + h * HDIM;  // strided rows
    float* ksb  = ks  + buf * 1024;
    float* kbsb = kbs + buf * 1024;
    float* vsb  = vs  + buf * 1024;
    float* aksb = aks + buf * 1024;
    #pragma unroll
    for (int st = 0; st < 8; ++st) {
      int i4 = (st * 32 + lane) * 4;          // 0,4,...,1020
      int r = i4 >> 6, c = i4 & 63;
      async_copy_b128(kp_g + i4, ksb + i4);
      async_copy_b128(kb_g + i4, kbsb + i4);
      async_copy_b128(v_g  + i4, vsb + i4);
      async_copy_b128(ak_g + r * DD + c, aksb + i4);
    }
  };

  issue_tile(0, 0);
  wait_asynccnt0();
  __syncthreads();

  v8f O0 = {}, O1 = {}, O2 = {}, O3 = {};
  float rowmax[8], rowsum[8];
  #pragma unroll
  for (int r = 0; r < 8; ++r) { rowmax[r] = -__builtin_inff(); rowsum[r] = 0.f; }
  const int hi8 = (lane >> 4) * 8;
  const int nlo = lane & 15;

  for (int kt = 0; kt < SS / 16; ++kt) {
    const int buf = kt & 1;
    const float* ksb  = ks  + buf * 1024;
    const float* kbsb = kbs + buf * 1024;
    const float* vsb  = vs  + buf * 1024;
    const float* aksb = aks + buf * 1024;
    if (kt + 1 < SS / 16) issue_tile((kt + 1) * 16, buf ^ 1);   // prefetch next

    // scores tile = Q@K'^T + Q2@K^T  (K=64 in 16 chunks of 4)
    v8f acc = {};
    #pragma unroll
    for (int kc = 0; kc < 16; ++kc) {
      v2f a1 = frag_rowmajor(qs,  lane, kc * 4, HDIM);
      v2f b1 = frag_rowmajor(ksb, lane, kc * 4, HDIM);   // B(k,n)=K'[n][k]
      acc = __builtin_amdgcn_wmma_f32_16x16x4_f32(false, a1, false, b1, (short)0, acc, false, false);
      v2f a2 = frag_rowmajor(qps,  lane, kc * 4, HDIM);
      v2f b2 = frag_rowmajor(kbsb, lane, kc * 4, HDIM);
      acc = __builtin_amdgcn_wmma_f32_16x16x4_f32(false, a2, false, b2, (short)0, acc, false, false);
    }

    // tanh rel-bias term + online softmax (row = 16 lanes of a half-wave)
    #pragma unroll
    for (int r = 0; r < 8; ++r) {
      int m = r + hi8;
      const float* aqr = aqs + m * HDIM;
      const float* akr = aksb + nlo * HDIM;
      float b3 = 0.f;
      for (int d = 0; d < HDIM; ++d) b3 += fast_tanh(akr[d] - aqr[d]) * rb[d];
      float s = acc[r] * INV_SQRT_HD + grh * b3;
      float tmax = s;
      #pragma unroll
      for (int off = 8; off > 0; off >>= 1) tmax = fmaxf(tmax, __shfl_xor(tmax, off, 16));
      float nm = fmaxf(rowmax[r], tmax);
      float p = __expf(s - nm);
      float psum = p;
      #pragma unroll
      for (int off = 8; off > 0; off >>= 1) psum += __shfl_xor(psum, off, 16);
      float corr = __expf(rowmax[r] - nm);
      rowsum[r] = rowsum[r] * corr + psum;
      rowmax[r] = nm;
      O0[r] *= corr; O1[r] *= corr; O2[r] *= corr; O3[r] *= corr;
      Ps[m * 16 + nlo] = p;
    }
    __syncthreads();

    // O += P @ V  (16x16 x 16x64) via WMMA, 4 N-blocks x 4 K-chunks
    #pragma unroll
    for (int kc = 0; kc < 4; ++kc) {
      v2f pa = frag_rowmajor(Ps, lane, kc * 4, 16);
      O0 = __builtin_amdgcn_wmma_f32_16x16x4_f32(false, pa, false, frag_colmajor(vsb, lane, kc * 4,  0, HDIM), (short)0, O0, false, false);
      O1 = __builtin_amdgcn_wmma_f32_16x16x4_f32(false, pa, false, frag_colmajor(vsb, lane, kc * 4, 16, HDIM), (short)0, O1, false, false);
      O2 = __builtin_amdgcn_wmma_f32_16x16x4_f32(false, pa, false, frag_colmajor(vsb, lane, kc * 4, 32, HDIM), (short)0, O2, false, false);
      O3 = __builtin_amdgcn_wmma_f32_16x16x4_f32(false, pa, false, frag_colmajor(vsb, lane, kc * 4, 48, HDIM), (short)0, O3, false, false);
    }
    wait_asynccnt0();   // next tile's async copies must have landed
    __syncthreads();
  }

  #pragma unroll
  for (int r = 0; r < 8; ++r) {
    int m = r + hi8;
    float invsum = 1.0f / rowsum[r];
    size_t o = ((size_t)(b * SS + q0 + m)) * DD + h * HDIM + nlo;
    gauge[o]      = O0[r] * invsum;
    gauge[o + 16] = O1[r] * invsum;
    gauge[o + 32] = O2[r] * invsum;
    gauge[o + 48] = O3[r] * invsum;
  }
}

// --------- Kernel 4: delta_v GEMM (WMMA) + gating + final scale -----------
__global__ void __launch_bounds__(32) final_kernel(
    const float* __restrict__ Av, const float* __restrict__ val_w,
    const float* __restrict__ gauge, const float* __restrict__ g_val,
    const float* __restrict__ out_scale, float* __restrict__ out) {
  __shared__ __align__(16) float As[16 * 64], Bs[16 * 64];
  const int lane = threadIdx.x;
  const int nt = blockIdx.x & (DD / 16 - 1);
  const int mt = blockIdx.x >> 5;
  const int m0 = mt * 16, n0 = nt * 16;
  v8f acc = {};
  for (int kc0 = 0; kc0 < DD; kc0 += 64) {
    #pragma unroll
    for (int st = 0; st < 8; ++st) {
      int i4 = (st * 32 + lane) * 4;
      int r = i4 >> 6, c = i4 & 63;
      async_copy_b128(Av    + (size_t)(m0 + r) * DD + kc0 + c, As + i4);
      async_copy_b128(val_w + (size_t)(n0 + r) * DD + kc0 + c, Bs + i4);  // B(k,n)=val_w[n][k]
    }
    wait_asynccnt0();
    __syncthreads();
    #pragma unroll
    for (int kc = 0; kc < 16; ++kc) {
      v2f a = frag_rowmajor(As, lane, kc * 4, 64);
      v2f b = frag_rowmajor(Bs, lane, kc * 4, 64);
      acc = __builtin_amdgcn_wmma_f32_16x16x4_f32(false, a, false, b, (short)0, acc, false, false);
    }
    __syncthreads();
  }
  float osc = fast_tanh(out_scale[0]);
  int n = n0 + (lane & 15);
  float gv = fast_tanh(g_val[n]);
  int hi8 = (lane >> 4) * 8;
  #pragma unroll
  for (int r = 0; r < 8; ++r) {
    size_t o = (size_t)(m0 + r + hi8) * DD + n;
    out[o] = osc * (gauge[o] + gv * acc[r]);
  }
}

extern "C" void kernel_launch(void* const* d_in, const int* in_sizes, int n_in,
                              void* d_out, int out_size, void* d_ws, size_t ws_size,
                              hipStream_t stream) {
  const float* hidden    = (const float*)d_in[0];
  const float* q_base    = (const float*)d_in[1];
  const float* k_base    = (const float*)d_in[2];
  const float* v_base    = (const float*)d_in[3];
  const float* ln_g      = (const float*)d_in[4];
  const float* ln_b      = (const float*)d_in[5];
  const float* fg_w1     = (const float*)d_in[6];
  const float* fg_w2     = (const float*)d_in[7];
  const float* aq_w      = (const float*)d_in[8];
  const float* ak_w      = (const float*)d_in[9];
  const float* val_w     = (const float*)d_in[10];
  const float* rel_bias  = (const float*)d_in[11];
  const float* g_attn    = (const float*)d_in[12];
  const float* g_rel     = (const float*)d_in[13];
  const float* g_val     = (const float*)d_in[14];
  const float* out_scale = (const float*)d_in[15];

  float* ws = (float*)d_ws;
  const size_t N1 = (size_t)BB * SS * DD;   // 524288
  float* Aq    = ws;
  float* Ak    = ws + N1;
  float* Av    = ws + 2 * N1;
  float* Kp    = ws + 3 * N1;
  float* Qp2   = ws + 4 * N1;
  float* gauge = ws + 5 * N1;

  prep_kernel<<<BB * SS, 256, 0, stream>>>(hidden, ln_g, ln_b, fg_w1, fg_w2, Aq, Ak, Av);
  headprep_kernel<<<(BB * HH * SS * HDIM) / 256, 256, 0, stream>>>(
      Aq, Ak, k_base, aq_w, ak_w, g_attn, Qp2, Kp);
  attn_kernel<<<BB * HH * (SS / 16), 32, 0, stream>>>(
      q_base, k_base, v_base, Kp, Qp2, Aq, Ak, rel_bias, g_rel, gauge);
  final_kernel<<<(BB * SS / 16) * (DD / 16), 32, 0, stream>>>(
      Av, val_w, gauge, g_val, out_scale, (float*)d_out);
}